// LinearRNN_17111149707714
// MI455X (gfx1250) — compile-verified
//
#include <hip/hip_runtime.h>

typedef float v2f __attribute__((ext_vector_type(2)));
typedef float v8f __attribute__((ext_vector_type(8)));
typedef unsigned int u32x4 __attribute__((ext_vector_type(4)));
typedef int i32x4 __attribute__((ext_vector_type(4)));
typedef int i32x8 __attribute__((ext_vector_type(8)));

#define B_   64
#define T_   1024
#define I_   512
#define H_   1024
#define NWG  64
#define PAD  4
#define SEQ_ROW 516   // 512 dwords + 2 dwords TDM pad per 256-dword interval

// Issue one TDM descriptor: DMA the 64x512 f32 tile seqs[:, t, :] into LDS,
// with 2-dword padding every 256 dwords (bank-conflict-free A reads).
__device__ __forceinline__ void tdm_prefetch_seq(const float* seqs, int t,
                                                 unsigned lds_base)
{
    unsigned long long ga = (unsigned long long)(uintptr_t)seqs
                          + (unsigned long long)t * (I_ * sizeof(float));
    u32x4 g0;
    g0.x = 1u;                                     // count=1, user mode, no gather
    g0.y = lds_base;                               // LDS dest byte address
    g0.z = (unsigned)ga;                           // global_addr[31:0]
    g0.w = (unsigned)((ga >> 32) & 0x01FFFFFFull)  // global_addr[56:32]
         | (2u << 30);                             // D# type = 2 ("image")
    i32x8 g1;
    g1[0] = (2 << 16)        // data_size = 4 bytes
          | (1 << 20)        // pad_enable
          | (7 << 22)        // pad_interval code 7 = every 256 dwords
          | (1 << 25);       // pad_amount  code 1 = 2 dwords
    g1[1] = (int)(I_ << 16); // tensor_dim0 = 512 (bits 79:48 -> [31:16])
    g1[2] = (int)(B_ << 16); // tensor_dim1 = 64  (bits 111:80 -> [31:16])
    g1[3] = (int)(I_ << 16); // tile_dim0  = 512  (bits 127:112)
    g1[4] = B_;              // tile_dim1 = 64, tile_dim2 = 0 (2D tile)
    g1[5] = (int)(T_ * I_);  // tensor_dim0_stride = T*I elements (batch row)
    g1[6] = 0;
    g1[7] = 0;
    i32x4 z4 = {0, 0, 0, 0};          // groups 2/3 unused for 2D
    i32x8 z8 = {0, 0, 0, 0, 0, 0, 0, 0};
    __builtin_amdgcn_tensor_load_to_lds(g0, g1, z4, z4, z8, 0);
}

// Persistent fused linear-RNN kernel.
// WG g owns output columns n0 = 16*g .. n0+15 for all 64 batch rows.
// Wave pair (2*mtile, 2*mtile+1) computes the 16x16 tile (rows 16*mtile..+15,
// cols n0..+15), splitting fused K = 512 (seqs via W_ih) + 1024 (h via W_hh)
// into two 768-wide halves reduced through LDS.
__global__ __launch_bounds__(256, 1) void rnn_persistent(
    const float* __restrict__ seqs, const int* __restrict__ lens,
    const float* __restrict__ W_ih, const float* __restrict__ b_ih,
    const float* __restrict__ W_hh, const float* __restrict__ b_hh,
    float* __restrict__ out, float* __restrict__ hA, float* __restrict__ hB,
    unsigned* __restrict__ sync_cnt)
{
    __shared__ float Whh_s[16][H_ + PAD];   // Whh_s[j][k] = W_hh[n0+j][k]
    __shared__ float Wih_s[16][I_ + PAD];   // Wih_s[j][k] = W_ih[n0+j][k]
    __shared__ float seq_s[SEQ_ROW * B_];   // TDM-staged seqs[:, t, :], padded rows
    __shared__ float red_s[4][16][17];      // cross-wave K reduction
    __shared__ float bias_s[16];
    __shared__ int   lens_s[B_];

    const int g     = blockIdx.x;
    const int n0    = g * 16;
    const int tid   = threadIdx.x;
    const int wave  = tid >> 5;
    const int lane  = tid & 31;
    const int mtile = wave >> 1;     // 0..3 -> batch rows 16*mtile..+15
    const int half  = wave & 1;      // K-split half
    const int m0    = mtile * 16;
    const int row   = lane & 15;     // A-row within tile / B-col within tile
    const int hi    = lane >> 4;     // lane group selects K pair within 4-step
    const int koff  = hi * 2;

    // ---- one-time staging: W slices, bias, lens into LDS ----
    for (int idx = tid; idx < 16 * H_; idx += 256) {
        int j = idx >> 10, k = idx & (H_ - 1);
        Whh_s[j][k] = W_hh[(size_t)(n0 + j) * H_ + k];
    }
    for (int idx = tid; idx < 16 * I_; idx += 256) {
        int j = idx >> 9, k = idx & (I_ - 1);
        Wih_s[j][k] = W_ih[(size_t)(n0 + j) * I_ + k];
    }
    if (tid < 16) bias_s[tid] = b_ih[n0 + tid] + b_hh[n0 + tid];
    if (tid < B_) lens_s[tid] = lens[tid];
    __syncthreads();

    const unsigned seq_lds = (unsigned)(uintptr_t)&seq_s[0];
    if (wave == 0) tdm_prefetch_seq(seqs, 0, seq_lds);   // tile for t = 0

    const int b_row = m0 + row;                 // batch row this lane feeds as A
    unsigned target = 0;

    for (int t = 0; t < T_; ++t) {
        const float* __restrict__ hprev = (t & 1) ? hB : hA;
        float* __restrict__ hnew        = (t & 1) ? hA : hB;

        // seqs tile for this step must be in LDS before anyone reads it.
        if (wave == 0) __builtin_amdgcn_s_wait_tensorcnt(0);
        __syncthreads();

        v8f c = {};
        if (half == 0) {
            // seqs_t contribution, K = 0..511 (A from TDM-staged LDS tile)
            const float* srow = &seq_s[SEQ_ROW * b_row];
            #pragma unroll 4
            for (int k = 0; k < 256; k += 4) {
                v2f a = *(const v2f*)(srow + k + koff);
                v2f b = *(const v2f*)(&Wih_s[row][k + koff]);
                c = __builtin_amdgcn_wmma_f32_16x16x4_f32(
                        false, a, false, b, (short)0, c, false, false);
            }
            #pragma unroll 4
            for (int k = 256; k < I_; k += 4) {       // +2: TDM pad at 256 dwords
                v2f a = *(const v2f*)(srow + 2 + k + koff);
                v2f b = *(const v2f*)(&Wih_s[row][k + koff]);
                c = __builtin_amdgcn_wmma_f32_16x16x4_f32(
                        false, a, false, b, (short)0, c, false, false);
            }
            // h contribution, K = 0..255
            const float* hptr = hprev + (size_t)b_row * H_;
            #pragma unroll 4
            for (int k = 0; k < 256; k += 4) {
                v2f a = *(const v2f*)(hptr + k + koff);
                v2f b = *(const v2f*)(&Whh_s[row][k + koff]);
                c = __builtin_amdgcn_wmma_f32_16x16x4_f32(
                        false, a, false, b, (short)0, c, false, false);
            }
        } else {
            // h contribution, K = 256..1023
            const float* hptr = hprev + (size_t)b_row * H_;
            #pragma unroll 4
            for (int k = 256; k < H_; k += 4) {
                v2f a = *(const v2f*)(hptr + k + koff);
                v2f b = *(const v2f*)(&Whh_s[row][k + koff]);
                c = __builtin_amdgcn_wmma_f32_16x16x4_f32(
                        false, a, false, b, (short)0, c, false, false);
            }
            #pragma unroll
            for (int v = 0; v < 8; ++v)
                red_s[mtile][v + hi * 8][row] = c[v];
        }
        __syncthreads();   // compute done: seq_s reads complete, red_s published

        // Overlap next step's seqs DMA with epilogue + device-wide barrier.
        if (wave == 0 && t + 1 < T_) tdm_prefetch_seq(seqs, t + 1, seq_lds);

        if (half == 0) {
            const int ncol = n0 + row;          // C/D: col = lane%16
            const float bv = bias_s[row];
            #pragma unroll
            for (int v = 0; v < 8; ++v) {
                const int m = m0 + v + hi * 8;  // C/D: row = VGPR idx (+8 hi lanes)
                float val = c[v] + red_s[mtile][v + hi * 8][row] + bv;
                hnew[(size_t)m * H_ + ncol] = val;
                if (lens_s[m] == t + 1)
                    out[(size_t)m * H_ + ncol] = val;
            }
        }

        // ---- device-wide barrier (monotonic counter, release/acquire) ----
        target += NWG;
        __syncthreads();
        if (tid == 0) {
            __hip_atomic_fetch_add(sync_cnt, 1u, __ATOMIC_ACQ_REL,
                                   __HIP_MEMORY_SCOPE_AGENT);
            while (__hip_atomic_load(sync_cnt, __ATOMIC_ACQUIRE,
                                     __HIP_MEMORY_SCOPE_AGENT) < target)
                __builtin_amdgcn_s_sleep(2);
        }
        __syncthreads();
    }
}

extern "C" void kernel_launch(void* const* d_in, const int* in_sizes, int n_in,
                              void* d_out, int out_size, void* d_ws, size_t ws_size,
                              hipStream_t stream) {
    const float* seqs = (const float*)d_in[0];
    const int*   lens = (const int*)  d_in[1];
    const float* W_ih = (const float*)d_in[2];
    const float* b_ih = (const float*)d_in[3];
    const float* W_hh = (const float*)d_in[4];
    const float* b_hh = (const float*)d_in[5];
    float* out = (float*)d_out;

    unsigned char* ws = (unsigned char*)d_ws;
    unsigned* cnt = (unsigned*)ws;                       // barrier counter
    float* hA = (float*)(ws + 256);                      // h ping
    float* hB = hA + (size_t)B_ * H_;                    // h pong

    // Zero barrier counter and hA (== h_{-1} = 0) every call; capture-legal.
    (void)hipMemsetAsync(ws, 0, 256 + (size_t)B_ * H_ * sizeof(float), stream);

    rnn_persistent<<<NWG, 256, 0, stream>>>(seqs, lens, W_ih, b_ih, W_hh, b_hh,
                                            out, hA, hB, cnt);
}